// StripDynamicConv2d_79731772883304
// MI455X (gfx1250) — compile-verified
//
#include <hip/hip_runtime.h>
#include <hip/hip_bf16.h>
#include <math.h>

// ---------------------------------------------------------------------------
// StripDynamicConv2d for MI455X (gfx1250, wave32)
//   B=16, C=256, H=W=96, G=4, K=3, RED=4
// Phase split:
//   k1: pooling (block means 32x32 -> 3x3, plus global mean)  [memory pass 1]
//   k2: projection MLP via V_WMMA_F32_16X16X4_F32             [matrix cores]
//   k3: softmax-over-G mixing of w_k / b_k                    [tiny]
//   k4: depthwise 3x3 conv, LDS-staged, float4 (b128) I/O     [memory pass 2]
// ---------------------------------------------------------------------------

typedef __attribute__((ext_vector_type(2))) float v2f;
typedef __attribute__((ext_vector_type(8))) float v8f;

#define NB   16
#define NC   256
#define NH   96
#define NW   96
#define NG   4
#define NKK  9           // 3x3
#define CR   64          // C / RED
#define CG   1024        // C * G
#define PLANE (NH*NW)    // 9216

// ---------------------------------------------------------------------------
// Kernel 1: per-(b,c) plane pooling.
// 288 threads = 9 waves; wave kk reduces pool block (kk/3, kk%3) of 32x32.
// Output layout: inmat[b][c][n], n in [0,16): n<9 = pooled, n==9 = gmean, rest 0.
// ---------------------------------------------------------------------------
__global__ __launch_bounds__(288)
void pool_kernel(const float* __restrict__ x, float* __restrict__ inmat)
{
    __shared__ float rsum[288];
    __shared__ float bsum[9];
    const int plane = blockIdx.x;          // b*256 + c
    const int t  = threadIdx.x;
    const int kk = t >> 5;                 // 0..8  (wave id)
    const int r  = t & 31;                 // lane
    const int kr = kk / 3, kc = kk % 3;

    const float4* px = (const float4*)(x + (size_t)plane * PLANE);
    // row = kr*32 + r; cols kc*32 .. kc*32+31  -> 8 float4
    const int base4 = (kr * 32 + r) * (NW / 4) + kc * 8;
    float s = 0.f;
#pragma unroll
    for (int i = 0; i < 8; ++i) {
        float4 v = px[base4 + i];
        s += v.x + v.y + v.z + v.w;
    }
    rsum[t] = s;
    __syncthreads();
    if (r == 0) {
        float bs = 0.f;
#pragma unroll
        for (int i = 0; i < 32; ++i) bs += rsum[kk * 32 + i];
        bsum[kk] = bs;
    }
    __syncthreads();
    if (t == 0) {
        float* dst = inmat + (size_t)plane * 16;
        float tot = 0.f;
#pragma unroll
        for (int i = 0; i < 9; ++i) { tot += bsum[i]; dst[i] = bsum[i] * (1.0f / 1024.0f); }
        dst[9] = tot * (1.0f / (float)PLANE);
#pragma unroll
        for (int i = 10; i < 16; ++i) dst[i] = 0.f;
    }
}

// ---------------------------------------------------------------------------
// Kernel 2: projection MLP with f32 WMMA.
//   h(64x16)    = GELU(BN(w1(64x256) @ in(256x16) + b1))
//   out2(1024x16) = w2(1024x64) @ h + b2
// One workgroup per batch b; 4 waves. K-loop step 4 via V_WMMA_F32_16X16X4_F32.
// A-frag (16x4 f32): lane L<16 -> row L, K={k,k+1}; lane>=16 -> row L-16, K={k+2,k+3}
// B-frag (4x16 f32): lane L<16 -> col L, K={k,k+1}; lane>=16 -> col L-16, K={k+2,k+3}
// C/D: VGPR v -> row v + 8*(lane>=16), col = lane&15
// ---------------------------------------------------------------------------
__global__ __launch_bounds__(128)
void proj_wmma_kernel(const float* __restrict__ w1, const float* __restrict__ b1,
                      const float* __restrict__ bn_g, const float* __restrict__ bn_b,
                      const float* __restrict__ bn_m, const float* __restrict__ bn_v,
                      const float* __restrict__ w2, const float* __restrict__ b2,
                      const float* __restrict__ inmat, float* __restrict__ out2)
{
    __shared__ float h_lds[CR * 16];
    const int b    = blockIdx.x;
    const int lane = threadIdx.x & 31;
    const int wave = threadIdx.x >> 5;
    const int n    = lane & 15;
    const int hi   = lane >> 4;            // 0: lanes 0-15, 1: lanes 16-31
    const int koff = hi * 2;
    const float* in_b = inmat + (size_t)b * (NC * 16);

    // ---- layer 1: h = w1 @ in ----
    {
        const int mt = wave;               // 4 m-tiles over 4 waves
        const int m  = mt * 16 + n;
        v8f acc = {};
        for (int k = 0; k < NC; k += 4) {
            v2f a = *(const v2f*)(w1 + m * NC + k + koff);
            v2f bb;
            bb.x = in_b[(k + koff    ) * 16 + n];
            bb.y = in_b[(k + koff + 1) * 16 + n];
            acc = __builtin_amdgcn_wmma_f32_16x16x4_f32(
                      false, a, false, bb, (short)0, acc, false, false);
        }
#pragma unroll
        for (int v = 0; v < 8; ++v) {
            const int row = mt * 16 + v + hi * 8;     // output channel of layer 1
            float val = acc[v] + b1[row];
            const float invstd = rsqrtf(bn_v[row] + 1e-5f);
            val = (val - bn_m[row]) * (bn_g[row] * invstd) + bn_b[row];
            val = 0.5f * val * (1.0f + erff(val * 0.70710678118654752f)); // exact GELU
            h_lds[row * 16 + n] = val;
        }
    }
    __syncthreads();

    // ---- layer 2: out2 = w2 @ h ----
    float* o_b = out2 + (size_t)b * (CG * 16);
    for (int mt = wave; mt < CG / 16; mt += 4) {       // 64 tiles / 4 waves
        const int m = mt * 16 + n;
        v8f acc = {};
#pragma unroll
        for (int k = 0; k < CR; k += 4) {
            v2f a = *(const v2f*)(w2 + m * CR + k + koff);
            v2f bb;
            bb.x = h_lds[(k + koff    ) * 16 + n];
            bb.y = h_lds[(k + koff + 1) * 16 + n];
            acc = __builtin_amdgcn_wmma_f32_16x16x4_f32(
                      false, a, false, bb, (short)0, acc, false, false);
        }
#pragma unroll
        for (int v = 0; v < 8; ++v) {
            const int row = mt * 16 + v + hi * 8;
            o_b[row * 16 + n] = acc[v] + b2[row];
        }
    }
}

// ---------------------------------------------------------------------------
// Kernel 3: softmax over G, mix static kernels/bias into dynamic ones.
//   wbuf[b][c][n] = sum_g softmax_g(out2[b][g*256+c][n]) * w_k[g][c][n], n<9
//   bbuf[b][c]    = sum_g softmax_g(out2[b][g*256+c][9]) * b_k[g][c]
// ---------------------------------------------------------------------------
__global__ __launch_bounds__(256)
void combine_kernel(const float* __restrict__ out2, const float* __restrict__ w_k,
                    const float* __restrict__ b_k, float* __restrict__ wbuf,
                    float* __restrict__ bbuf)
{
    const int idx = blockIdx.x * 256 + threadIdx.x;    // b*256 + c
    if (idx >= NB * NC) return;
    const int b = idx >> 8, c = idx & 255;
    const float* o = out2 + (size_t)b * (CG * 16);

#pragma unroll
    for (int n = 0; n < 10; ++n) {
        float l0 = o[(0 * NC + c) * 16 + n];
        float l1 = o[(1 * NC + c) * 16 + n];
        float l2 = o[(2 * NC + c) * 16 + n];
        float l3 = o[(3 * NC + c) * 16 + n];
        float mx = fmaxf(fmaxf(l0, l1), fmaxf(l2, l3));
        float e0 = expf(l0 - mx), e1 = expf(l1 - mx);
        float e2 = expf(l2 - mx), e3 = expf(l3 - mx);
        float inv = 1.0f / (e0 + e1 + e2 + e3);
        if (n < 9) {
            float w = (e0 * w_k[(0 * NC + c) * NKK + n] +
                       e1 * w_k[(1 * NC + c) * NKK + n] +
                       e2 * w_k[(2 * NC + c) * NKK + n] +
                       e3 * w_k[(3 * NC + c) * NKK + n]) * inv;
            wbuf[(size_t)idx * NKK + n] = w;
        } else {
            float bia = (e0 * b_k[0 * NC + c] + e1 * b_k[1 * NC + c] +
                         e2 * b_k[2 * NC + c] + e3 * b_k[3 * NC + c]) * inv;
            bbuf[idx] = bia;
        }
    }
}

// ---------------------------------------------------------------------------
// Kernel 4: depthwise 3x3 SAME conv, the bandwidth-bound pass.
// One WG per (b,c) plane: stage 96x96 (36 KB) in LDS via b128 loads,
// each thread produces 9 float4 outputs (b128 stores). ~302 MB total traffic.
// ---------------------------------------------------------------------------
__global__ __launch_bounds__(256)
void dwconv_kernel(const float* __restrict__ x, const float* __restrict__ wbuf,
                   const float* __restrict__ bbuf, float* __restrict__ out)
{
    __shared__ float tile[PLANE];
    const int plane = blockIdx.x;          // b*256 + c
    const int t = threadIdx.x;

    const float4* px = (const float4*)(x + (size_t)plane * PLANE);
    float4* pt = (float4*)tile;
#pragma unroll
    for (int i = 0; i < 9; ++i) pt[t + i * 256] = px[t + i * 256];

    float w[9];
    const float* wp = wbuf + (size_t)plane * NKK;
#pragma unroll
    for (int i = 0; i < 9; ++i) w[i] = wp[i];
    const float bias = bbuf[plane];
    __syncthreads();

    float4* po = (float4*)(out + (size_t)plane * PLANE);
#pragma unroll 1
    for (int i = 0; i < 9; ++i) {
        const int idx4 = t + i * 256;          // 0..2303
        const int y  = idx4 / (NW / 4);
        const int x0 = (idx4 % (NW / 4)) * 4;
        float4 res;
        float* rp = &res.x;
#pragma unroll
        for (int j = 0; j < 4; ++j) {
            const int xc = x0 + j;
            float acc = bias;
#pragma unroll
            for (int dy = -1; dy <= 1; ++dy) {
                const int yy = y + dy;
                if (yy < 0 || yy >= NH) continue;
#pragma unroll
                for (int dx = -1; dx <= 1; ++dx) {
                    const int xx = xc + dx;
                    if (xx < 0 || xx >= NW) continue;
                    acc += w[(dy + 1) * 3 + (dx + 1)] * tile[yy * NW + xx];
                }
            }
            rp[j] = acc;
        }
        po[idx4] = res;
    }
}

// ---------------------------------------------------------------------------
// Launch
// ---------------------------------------------------------------------------
extern "C" void kernel_launch(void* const* d_in, const int* in_sizes, int n_in,
                              void* d_out, int out_size, void* d_ws, size_t ws_size,
                              hipStream_t stream) {
    const float* x    = (const float*)d_in[0];
    const float* w_k  = (const float*)d_in[1];
    const float* b_k  = (const float*)d_in[2];
    const float* w1   = (const float*)d_in[3];
    const float* b1   = (const float*)d_in[4];
    const float* bn_g = (const float*)d_in[5];
    const float* bn_b = (const float*)d_in[6];
    const float* bn_m = (const float*)d_in[7];
    const float* bn_v = (const float*)d_in[8];
    const float* w2   = (const float*)d_in[9];
    const float* b2   = (const float*)d_in[10];
    float* out = (float*)d_out;

    // workspace layout (floats): ~1.44 MB total
    float* ws    = (float*)d_ws;
    float* inmat = ws;                               // 16*256*16   = 65536
    float* out2  = inmat + NB * NC * 16;             // 16*1024*16  = 262144
    float* wbuf  = out2 + NB * CG * 16;              // 16*256*9    = 36864
    float* bbuf  = wbuf + NB * NC * NKK;             // 16*256      = 4096

    pool_kernel<<<NB * NC, 288, 0, stream>>>(x, inmat);
    proj_wmma_kernel<<<NB, 128, 0, stream>>>(w1, b1, bn_g, bn_b, bn_m, bn_v,
                                             w2, b2, inmat, out2);
    combine_kernel<<<(NB * NC + 255) / 256, 256, 0, stream>>>(out2, w_k, b_k, wbuf, bbuf);
    dwconv_kernel<<<NB * NC, 256, 0, stream>>>(x, wbuf, bbuf, out);
}